// GMF_70866960384287
// MI455X (gfx1250) — compile-verified
//
#include <hip/hip_runtime.h>

typedef __attribute__((ext_vector_type(2))) float v2f;
typedef __attribute__((ext_vector_type(8))) float v8f;

#define N_USERS   1000000
#define N_ITEMS   100000
#define N_GENRES  32
#define KDIM      64
#define QSTRIDE   (N_ITEMS + N_GENRES)
#define QG_STRIDE 34   // padded LDS row stride (banks spread for ds_load_b64)

__global__ __launch_bounds__(256) void gmf_fused_kernel(
    const int*   __restrict__ user_ids,   // [B]
    const int*   __restrict__ item_ids,   // [B]
    const float* __restrict__ genres,     // [B, 32]
    const float* __restrict__ P,          // [64, N_USERS]
    const float* __restrict__ Q,          // [64, QSTRIDE]
    float*       __restrict__ out,        // [B]
    int B)
{
    // Stage Qg = Q[:, N_ITEMS:] (64 x 32 f32, 8KB) into LDS once per block.
    __shared__ float qg[KDIM * QG_STRIDE];
    const int tid  = threadIdx.x;
    const int lane = tid & 31;
    const int wave = tid >> 5;

    #pragma unroll
    for (int i = 0; i < (KDIM * N_GENRES) / 256; ++i) {
        int e = tid + i * 256;
        int j = e >> 5;        // 0..63 (latent dim)
        int g = e & 31;        // 0..31 (genre)
        qg[j * QG_STRIDE + g] = Q[(size_t)j * QSTRIDE + N_ITEMS + g];
    }
    __syncthreads();

    const int tile = blockIdx.x * 8 + wave;   // one 16-row batch tile per wave
    const int b0   = tile * 16;
    const int n    = lane & 15;               // row (A) / col (B,C) within tile
    const int hi   = lane >> 4;               // K-pair select for A/B frags

    // ---- dense genre GEMM: q_genre[16x64] = genres[16x32] @ Qg^T[32x64] ----
    // via V_WMMA_F32_16X16X4_F32, 8 K-steps x 4 N-tiles.
    v8f acc[4] = {};
    {
        int arow = b0 + n;
        if (arow > B - 1) arow = B - 1;       // tail-safe (values unused)
        const float* gptr = genres + (size_t)arow * N_GENRES + 2 * hi;
        #pragma unroll
        for (int kk = 0; kk < 8; ++kk) {
            v2f a = *(const v2f*)(gptr + 4 * kk);          // A[m][4kk+2hi, +1]
            #pragma unroll
            for (int t = 0; t < 4; ++t) {
                v2f bf = *(const v2f*)&qg[(t * 16 + n) * QG_STRIDE + 4 * kk + 2 * hi];
                acc[t] = __builtin_amdgcn_wmma_f32_16x16x4_f32(
                    false, a, false, bf, (short)0, acc[t], false, false);
            }
        }
    }

    // ---- fused random gathers + dot-product reduction ----
    // C layout: acc[t][v] @ lane L = q_genre[row = v + 8*hi][j = 16t + (L&15)]
    #pragma unroll
    for (int v = 0; v < 8; ++v) {
        int row  = b0 + v + 8 * hi;
        int rowc = row < B ? row : B - 1;
        const int uid = user_ids[rowc];
        const int iid = item_ids[rowc];
        float partial = 0.f;
        #pragma unroll
        for (int t = 0; t < 4; ++t) {
            const int j = t * 16 + n;
            // P is 256MB, touched once: non-temporal so it doesn't thrash L2.
            float pu = __builtin_nontemporal_load(P + (size_t)j * N_USERS + uid);
            float qi = Q[(size_t)j * QSTRIDE + iid];       // 25MB: L2-resident
            partial += pu * (qi + acc[t][v]);
        }
        // reduce the 16 lanes sharing this row (j-dimension)
        #pragma unroll
        for (int off = 8; off; off >>= 1)
            partial += __shfl_xor(partial, off, 16);
        if (n == 0 && row < B)
            out[row] = partial;
    }
}

extern "C" void kernel_launch(void* const* d_in, const int* in_sizes, int n_in,
                              void* d_out, int out_size, void* d_ws, size_t ws_size,
                              hipStream_t stream) {
    const int*   user_ids = (const int*)d_in[0];
    const int*   item_ids = (const int*)d_in[1];
    const float* genres   = (const float*)d_in[2];
    const float* P        = (const float*)d_in[3];
    const float* Q        = (const float*)d_in[4];
    float* out = (float*)d_out;

    const int B      = in_sizes[0];
    const int tiles  = (B + 15) / 16;
    const int blocks = (tiles + 7) / 8;
    gmf_fused_kernel<<<blocks, 256, 0, stream>>>(user_ids, item_ids, genres, P, Q, out, B);
}